// WDECT_RBFSVM_15942918603131
// MI455X (gfx1250) — compile-verified
//
#include <hip/hip_runtime.h>
#include <math.h>

#define NB 64
#define NN 65536
#define NE 196608
#define NF 131072
#define ND 32
#define NT 32
#define RFF 2048
#define NC 10
#define SC 100.0f

typedef float v2f __attribute__((ext_vector_type(2)));
typedef float v8f __attribute__((ext_vector_type(8)));

// ---------------- nh = (x * w) @ v : [N, ND] ----------------
__global__ void k_nh(const float* __restrict__ x, const float* __restrict__ nw,
                     const float* __restrict__ v, float* __restrict__ nh) {
  int n = blockIdx.x * blockDim.x + threadIdx.x;
  if (n >= NN) return;
  float w  = nw[n];
  float x0 = x[n * 3 + 0] * w;
  float x1 = x[n * 3 + 1] * w;
  float x2 = x[n * 3 + 2] * w;
#pragma unroll
  for (int d = 0; d < ND; ++d)
    nh[n * ND + d] = x0 * v[d] + x1 * v[ND + d] + x2 * v[2 * ND + d];
}

// ---------------- zero ecc + counters ----------------
__global__ void k_zero(float* __restrict__ ecc, int* __restrict__ cnt, int ncnt) {
  int i = blockIdx.x * blockDim.x + threadIdx.x;
  if (i < NB * ND * NT) ecc[i] = 0.0f;
  if (i < ncnt) cnt[i] = 0;
}

// ---------------- bucket by graph: count / scan / scatter ----------------
__global__ void k_count(const int* __restrict__ owner, const int* __restrict__ batch,
                        int count, int* __restrict__ cnt) {
  int i = blockIdx.x * blockDim.x + threadIdx.x;
  if (i >= count) return;
  int node = owner ? owner[i] : i;
  atomicAdd(&cnt[batch[node]], 1);
}

__global__ void k_scan(const int* __restrict__ cE, const int* __restrict__ cF,
                       const int* __restrict__ cN,
                       int* __restrict__ oE, int* __restrict__ oF, int* __restrict__ oN,
                       int* __restrict__ uE, int* __restrict__ uF, int* __restrict__ uN) {
  if (blockIdx.x != 0 || threadIdx.x != 0) return;
  int s = 0;
  for (int b = 0; b < NB; ++b) { oE[b] = s; uE[b] = s; s += cE[b]; } oE[NB] = s;
  s = 0;
  for (int b = 0; b < NB; ++b) { oF[b] = s; uF[b] = s; s += cF[b]; } oF[NB] = s;
  s = 0;
  for (int b = 0; b < NB; ++b) { oN[b] = s; uN[b] = s; s += cN[b]; } oN[NB] = s;
}

__global__ void k_scatter(const int* __restrict__ owner, const int* __restrict__ batch,
                          int count, int* __restrict__ cur, int* __restrict__ perm) {
  int i = blockIdx.x * blockDim.x + threadIdx.x;
  if (i >= count) return;
  int node = owner ? owner[i] : i;
  int pos = atomicAdd(&cur[batch[node]], 1);
  perm[pos] = i;
}

// ---------------- ECC accumulation ----------------
// One block per (graph, chunk); thread = (d, t) pair; register accumulation;
// exactly one global atomic per thread at the end.
template <int MODE>
__global__ void k_ecc(const float* __restrict__ nh,
                      const int* __restrict__ perm, const int* __restrict__ offs,
                      const int* __restrict__ ei, const float* __restrict__ ew,
                      const int* __restrict__ fi, const float* __restrict__ fw,
                      const float* __restrict__ lin, float sign,
                      float* __restrict__ ecc) {
  int b = blockIdx.x;
  int d = threadIdx.x >> 5;
  int t = threadIdx.x & 31;
  float l = lin[t];
  float acc = 0.0f;
  int lo = offs[b], hi = offs[b + 1];
  for (int i = lo + (int)blockIdx.y; i < hi; i += (int)gridDim.y) {
    int it = perm[i];
    float h;
    if (MODE == 0) {
      h = nh[it * ND + d];
    } else if (MODE == 1) {
      int e0 = ei[it], e1 = ei[NE + it];
      h = fmaxf(nh[e0 * ND + d], nh[e1 * ND + d]) * ew[it];
    } else {
      int f0 = fi[it], f1 = fi[NF + it], f2 = fi[2 * NF + it];
      h = fmaxf(fmaxf(nh[f0 * ND + d], nh[f1 * ND + d]), nh[f2 * ND + d]) * fw[it];
    }
    float z = SC * (l - h);
    acc += __builtin_amdgcn_rcpf(1.0f + __expf(-z));  // sigmoid(z)
  }
  atomicAdd(&ecc[b * (ND * NT) + d * NT + t], sign * acc);
}

// ---------------- copy flat = ecc into output ----------------
__global__ void k_copyflat(const float* __restrict__ ecc, float* __restrict__ out) {
  int i = blockIdx.x * blockDim.x + threadIdx.x;
  if (i < NB * ND * NT) out[NB * NC + i] = ecc[i];
}

// ---------------- phi = (1/32) * cos(flat @ W_rff + b_rff) via f32 WMMA ----------
// One wave computes a 16x16 tile of [64 x 2048]; K=1024 in steps of 4.
// A layout (16x4 f32, ISA 7.12.2): lanes 0-15: M=lane, K={k0,k0+1}; lanes 16-31: K={k0+2,k0+3}.
// B layout (4x16):                 VGPR0: K=k0+2*half; VGPR1: K=k0+2*half+1; N=lane&15.
// C/D layout: VGPR r: M = r + 8*(lane>=16), N = lane&15.
__global__ void k_rff(const float* __restrict__ flat, const float* __restrict__ W,
                      const float* __restrict__ brff, float* __restrict__ phi) {
  int wid  = (blockIdx.x * blockDim.x + threadIdx.x) >> 5;  // 0..511
  int lane = threadIdx.x & 31;
  int half = lane >> 4;
  int l15  = lane & 15;
  int tm = (wid & 3) * 16;     // 4 M tiles
  int tn = (wid >> 2) * 16;    // 128 N tiles
  int arow = tm + l15;
  int bcol = tn + l15;
  const float* aptr = flat + arow * (ND * NT) + 2 * half;
  const float* bptr = W + (2 * half) * RFF + bcol;
  v8f c = {0.f, 0.f, 0.f, 0.f, 0.f, 0.f, 0.f, 0.f};
#pragma unroll 4
  for (int k0 = 0; k0 < ND * NT; k0 += 4) {
    v2f a;
    a.x = aptr[k0];
    a.y = aptr[k0 + 1];
    v2f b;
    b.x = bptr[(size_t)k0 * RFF];
    b.y = bptr[(size_t)(k0 + 1) * RFF];
    c = __builtin_amdgcn_wmma_f32_16x16x4_f32(
        /*neg_a=*/false, a, /*neg_b=*/false, b,
        /*c_mod=*/(short)0, c, /*reuse_a=*/false, /*reuse_b=*/false);
  }
  const float amp = 0.03125f;  // sqrt(2/RFF) = 1/32
  float bb = brff[bcol];
#pragma unroll
  for (int r = 0; r < 8; ++r) {
    int m = tm + r + 8 * half;
    phi[m * RFF + bcol] = amp * __cosf(c[r] + bb);
  }
}

// ---------------- logits = phi @ W_lin + b_lin ----------------
__global__ void k_logits(const float* __restrict__ phi, const float* __restrict__ Wl,
                         const float* __restrict__ bl, float* __restrict__ out) {
  int b = blockIdx.x;         // graph
  int lane = threadIdx.x;     // 32 lanes (wave32)
  for (int c = 0; c < NC; ++c) {
    float acc = 0.0f;
    for (int n = lane; n < RFF; n += 32)
      acc += phi[b * RFF + n] * Wl[n * NC + c];
#pragma unroll
    for (int off = 16; off; off >>= 1)
      acc += __shfl_down(acc, off, 32);
    if (lane == 0) out[b * NC + c] = acc + bl[c];
  }
}

extern "C" void kernel_launch(void* const* d_in, const int* in_sizes, int n_in,
                              void* d_out, int out_size, void* d_ws, size_t ws_size,
                              hipStream_t stream) {
  const float* x     = (const float*)d_in[0];
  const float* nw    = (const float*)d_in[1];
  const int*   ei    = (const int*)  d_in[2];
  const float* ew    = (const float*)d_in[3];
  const int*   fi    = (const int*)  d_in[4];
  const float* fw    = (const float*)d_in[5];
  const int*   batch = (const int*)  d_in[6];
  const float* v     = (const float*)d_in[7];
  const float* lin   = (const float*)d_in[8];
  const float* Wr    = (const float*)d_in[9];
  const float* br    = (const float*)d_in[10];
  const float* Wl    = (const float*)d_in[11];
  const float* bl    = (const float*)d_in[12];
  float* out = (float*)d_out;

  char* ws = (char*)d_ws;
  float* nh  = (float*)ws;                                   // 8 MB
  float* ecc = (float*)(ws + (size_t)NN * ND * 4);           // 256 KB
  float* phi = (float*)(ws + (size_t)NN * ND * 4 + NB * ND * NT * 4);  // 512 KB
  int* ip = (int*)(ws + (size_t)NN * ND * 4 + NB * ND * NT * 4 + (size_t)NB * RFF * 4);
  int* cntE = ip;        int* cntF = ip + 64;  int* cntN = ip + 128;
  int* offE = ip + 192;  int* offF = ip + 257; int* offN = ip + 322;  // 65 each
  int* curE = ip + 387;  int* curF = ip + 451; int* curN = ip + 515;
  int* permE = ip + 579;
  int* permF = permE + NE;
  int* permN = permF + NF;

  // zero ecc + the 3 count arrays
  k_zero<<<(NB * ND * NT + 255) / 256, 256, 0, stream>>>(ecc, cntE, 192);
  // node heights
  k_nh<<<(NN + 255) / 256, 256, 0, stream>>>(x, nw, v, nh);
  // bucket by graph
  k_count<<<(NE + 255) / 256, 256, 0, stream>>>(ei, batch, NE, cntE);
  k_count<<<(NF + 255) / 256, 256, 0, stream>>>(fi, batch, NF, cntF);
  k_count<<<(NN + 255) / 256, 256, 0, stream>>>((const int*)nullptr, batch, NN, cntN);
  k_scan<<<1, 1, 0, stream>>>(cntE, cntF, cntN, offE, offF, offN, curE, curF, curN);
  k_scatter<<<(NE + 255) / 256, 256, 0, stream>>>(ei, batch, NE, curE, permE);
  k_scatter<<<(NF + 255) / 256, 256, 0, stream>>>(fi, batch, NF, curF, permF);
  k_scatter<<<(NN + 255) / 256, 256, 0, stream>>>((const int*)nullptr, batch, NN, curN, permN);
  // ECC: nodes (+), edges (-), faces (+)
  dim3 eg(NB, 4);
  k_ecc<0><<<eg, 1024, 0, stream>>>(nh, permN, offN, ei, ew, fi, fw, lin,  1.0f, ecc);
  k_ecc<1><<<eg, 1024, 0, stream>>>(nh, permE, offE, ei, ew, fi, fw, lin, -1.0f, ecc);
  k_ecc<2><<<eg, 1024, 0, stream>>>(nh, permF, offF, ei, ew, fi, fw, lin,  1.0f, ecc);
  // outputs
  k_copyflat<<<(NB * ND * NT + 255) / 256, 256, 0, stream>>>(ecc, out);
  k_rff<<<64, 256, 0, stream>>>(ecc, Wr, br, phi);     // 512 waves, one 16x16 tile each
  k_logits<<<NB, 32, 0, stream>>>(phi, Wl, bl, out);
}